// ProbeMLP_85194971283916
// MI455X (gfx1250) — compile-verified
//
#include <hip/hip_runtime.h>
#include <hip/hip_bf16.h>

typedef float v2f __attribute__((ext_vector_type(2)));
typedef float v8f __attribute__((ext_vector_type(8)));

#define BB   16
#define NN   8192
#define DD   256
#define HH   128
#define NPAR 64
#define KPP  32
#define MTOT (BB * NN)        // 131072 rows
#define NSEL (NPAR * KPP)     // 2048 selected per batch

// ---------------------------------------------------------------------------
// Kernel 0: repack W1 [D,H] row-major into WMMA B-fragment order.
// For k-step ks (K = 4*ks) and n-tile nt, record = 32 lanes x float2:
//   lane l: khalf = l>>4, ncol = nt*16 + (l&15)
//   .x = W1[4*ks + 2*khalf    ][ncol]
//   .y = W1[4*ks + 2*khalf + 1][ncol]
// One float2 per thread; 64*8*32 = 16384 records. Makes every B-fragment
// load in the GEMM a single coalesced global_load_b64 (256 B / wave).
// ---------------------------------------------------------------------------
__global__ void __launch_bounds__(256)
repack_W1_kernel(const float* __restrict__ W1, float* __restrict__ Wp)
{
    const int t     = blockIdx.x * 256 + threadIdx.x;   // 0..16383
    const int lane  = t & 31;
    const int nt    = (t >> 5) & 7;
    const int ks    = t >> 8;                           // 0..63
    const int khalf = lane >> 4;
    const int ncol  = nt * 16 + (lane & 15);
    const int ka    = ks * 4 + 2 * khalf;
    v2f v;
    v.x = W1[ka * HH + ncol];
    v.y = W1[(ka + 1) * HH + ncol];
    ((v2f*)Wp)[t] = v;
}

// ---------------------------------------------------------------------------
// One K-step: 1 A-fragment load per M-tile + 8 coalesced B-fragment loads,
// 16 V_WMMA_F32_16X16X4_F32.
// ---------------------------------------------------------------------------
__device__ __forceinline__ void
kstep(int ks, int ka,
      const float* __restrict__ Arow0, const float* __restrict__ Arow1,
      const v2f* __restrict__ Bfrag, int lane, v8f* acc0, v8f* acc1)
{
    v2f a0, a1;
    a0.x = Arow0[ka]; a0.y = Arow0[ka + 1];
    a1.x = Arow1[ka]; a1.y = Arow1[ka + 1];
#pragma unroll
    for (int nt = 0; nt < 8; ++nt) {
        const v2f b = Bfrag[(ks * 8 + nt) * 32 + lane];   // coalesced b64
        acc0[nt] = __builtin_amdgcn_wmma_f32_16x16x4_f32(
            false, a0, false, b, (short)0, acc0[nt], false, false);
        acc1[nt] = __builtin_amdgcn_wmma_f32_16x16x4_f32(
            false, a1, false, b, (short)0, acc1[nt], false, false);
    }
}

// ---------------------------------------------------------------------------
// Fused epilogue for one 16-row C tile:
// hidden = relu(acc + b1); partial = hidden * W2; 16-lane reduce -> logits.
// C fragment: acc[nt][r] = element (M = r + 8*khalf, N = nt*16 + mrow).
// ---------------------------------------------------------------------------
__device__ __forceinline__ void
logit_epilogue(const v8f* acc,
               const float* __restrict__ b1, const float* __restrict__ W2,
               float bias2, const float* __restrict__ gumbel,
               float* __restrict__ logits_out, float* __restrict__ z_out,
               long rowBase, int mrow, int khalf)
{
    float sum[8];
#pragma unroll
    for (int r = 0; r < 8; ++r) sum[r] = 0.0f;

#pragma unroll
    for (int nt = 0; nt < 8; ++nt) {
        const int ncol = nt * 16 + mrow;
        const float bias = b1[ncol];
        const float w2   = W2[ncol];
#pragma unroll
        for (int r = 0; r < 8; ++r) {
            float h = acc[nt][r] + bias;
            h = h > 0.0f ? h : 0.0f;
            sum[r] += h * w2;
        }
    }
    // xor masks < 16 keep lanes inside their K-half group
#pragma unroll
    for (int off = 8; off >= 1; off >>= 1) {
#pragma unroll
        for (int r = 0; r < 8; ++r)
            sum[r] += __shfl_xor(sum[r], off, 32);
    }
#pragma unroll
    for (int r = 0; r < 8; ++r) {
        if (mrow == r) {
            const long m   = rowBase + r + 8 * khalf;   // = b*N + n
            const float lg = sum[r] + bias2;
            logits_out[m] = lg;
            z_out[m]      = lg + gumbel[m];             // softmax monotone:
        }                                               // rank on z directly
    }
}

// ---------------------------------------------------------------------------
// Kernel 1: fused MLP logits via V_WMMA_F32_16X16X4_F32.
// One wave owns TWO 16-row M-tiles (128 acc VGPRs) so each coalesced
// B-fragment load feeds 2 WMMAs. K-loop tiled by 4 so the A-stream prefetch
// (global_prefetch_b8) fires once per 16-float chunk with no inner branch;
// the final 8 k-steps run prefetch-free.
// ---------------------------------------------------------------------------
__global__ void __launch_bounds__(256)
mlp_logits_kernel(const float* __restrict__ features,
                  const float* __restrict__ Wp,       // repacked W1 fragments
                  const float* __restrict__ b1,
                  const float* __restrict__ W2,
                  const float* __restrict__ b2,
                  const float* __restrict__ gumbel,
                  float* __restrict__ logits_out,
                  float* __restrict__ z_out)
{
    const int lane = threadIdx.x & 31;
    const int wib  = threadIdx.x >> 5;
    const int pair = blockIdx.x * (blockDim.x >> 5) + wib;  // 32-row pair id
    const long rowBase0 = (long)pair * 32;
    const long rowBase1 = rowBase0 + 16;

    const int mrow  = lane & 15;       // M row within tile (A) / N col (B,C)
    const int khalf = lane >> 4;       // K-pair selector

    const float* __restrict__ Arow0 = features + (rowBase0 + mrow) * DD;
    const float* __restrict__ Arow1 = features + (rowBase1 + mrow) * DD;
    const v2f*   __restrict__ Bfrag = (const v2f*)Wp;       // [64][8][32]

    v8f acc0[8] = {};
    v8f acc1[8] = {};

    // steps 0..55: one prefetch pair per 4 k-steps (64 floats = 256 B ahead)
    for (int ks4 = 0; ks4 < 56; ks4 += 4) {
        const int kbase = ks4 * 4;
        __builtin_prefetch(Arow0 + kbase + 64, 0, 1);
        __builtin_prefetch(Arow1 + kbase + 64, 0, 1);
#pragma unroll
        for (int j = 0; j < 4; ++j) {
            const int ks = ks4 + j;
            kstep(ks, ks * 4 + 2 * khalf, Arow0, Arow1, Bfrag, lane,
                  acc0, acc1);
        }
    }
    // steps 56..63: tail, no prefetch
#pragma unroll
    for (int j = 0; j < 8; ++j) {
        const int ks = 56 + j;
        kstep(ks, ks * 4 + 2 * khalf, Arow0, Arow1, Bfrag, lane, acc0, acc1);
    }

    const float bias2 = b2[0];
    logit_epilogue(acc0, b1, W2, bias2, gumbel, logits_out, z_out,
                   rowBase0, mrow, khalf);
    logit_epilogue(acc1, b1, W2, bias2, gumbel, logits_out, z_out,
                   rowBase1, mrow, khalf);
}

// ---------------------------------------------------------------------------
// Kernel 2: per-(batch, parcel) top-32 of the 128 candidates {p, p+64, ...}.
// One wave per pair; 4 candidates/lane; 32 rounds of warp-argmax with
// JAX's lower-index tie-break. Emits indices in descending-value order.
// ---------------------------------------------------------------------------
__global__ void __launch_bounds__(256)
topk_kernel(const float* __restrict__ z,
            int*   __restrict__ selIdx,
            float* __restrict__ idx_out)
{
    const int wave = (int)((blockIdx.x * blockDim.x + threadIdx.x) >> 5);
    const int lane = threadIdx.x & 31;
    const int b = wave >> 6;       // / NPAR
    const int p = wave & 63;       // % NPAR
    const float* __restrict__ zb = z + (long)b * NN;

    float val[4];
    int   idx[4];
#pragma unroll
    for (int j = 0; j < 4; ++j) {
        const int n = p + NPAR * (lane + 32 * j);
        val[j] = zb[n];
        idx[j] = n;
    }

    for (int k = 0; k < KPP; ++k) {
        float bv = val[0]; int bi = idx[0]; int bj = 0;
#pragma unroll
        for (int j = 1; j < 4; ++j) {
            if (val[j] > bv || (val[j] == bv && idx[j] < bi)) {
                bv = val[j]; bi = idx[j]; bj = j;
            }
        }
        float wv = bv; int wi = bi;
#pragma unroll
        for (int off = 16; off >= 1; off >>= 1) {
            const float ov = __shfl_xor(wv, off, 32);
            const int   oi = __shfl_xor(wi, off, 32);
            if (ov > wv || (ov == wv && oi < wi)) { wv = ov; wi = oi; }
        }
        if (lane == 0) {
            const int pos = b * NSEL + p * KPP + k;    // parcel-major layout
            selIdx[pos]  = wi;
            idx_out[pos] = (float)wi;
        }
        if (bi == wi) val[bj] = -__builtin_inff();     // retire the winner
    }
}

// ---------------------------------------------------------------------------
// Kernel 3: gather. selected_patches[b, s, :] = patches[b, selIdx[b,s], :].
// (mask_grad == final_mask == 1.0 exactly at selected rows.)
// ---------------------------------------------------------------------------
__global__ void __launch_bounds__(256)
gather_kernel(const float* __restrict__ patches,
              const int*   __restrict__ selIdx,
              float* __restrict__ sel_out)
{
    const int wave = (int)((blockIdx.x * blockDim.x + threadIdx.x) >> 5);
    const int lane = threadIdx.x & 31;
    const int b = wave >> 11;      // / NSEL
    const int s = wave & 2047;     // % NSEL
    const int n = selIdx[b * NSEL + s];

    const float4* __restrict__ src =
        (const float4*)(patches + ((long)b * NN + n) * DD);
    float4* __restrict__ dst =
        (float4*)(sel_out + ((long)b * NSEL + s) * DD);

    dst[lane]      = src[lane];
    dst[lane + 32] = src[lane + 32];
}

// ---------------------------------------------------------------------------
extern "C" void kernel_launch(void* const* d_in, const int* in_sizes, int n_in,
                              void* d_out, int out_size, void* d_ws, size_t ws_size,
                              hipStream_t stream)
{
    const float* patches  = (const float*)d_in[0];
    const float* features = (const float*)d_in[1];
    const float* W1       = (const float*)d_in[2];
    const float* b1       = (const float*)d_in[3];
    const float* W2       = (const float*)d_in[4];
    const float* b2       = (const float*)d_in[5];
    const float* gumbel   = (const float*)d_in[6];
    // d_in[7] (lookup) is arange(N) % 64 by construction; parcels are implicit.

    // d_out layout (reference return order, flattened):
    float* sel_out = (float*)d_out;                       // [B, 2048, 256]
    float* idx_out = sel_out + (long)BB * NSEL * DD;      // [B, 2048]
    float* log_out = idx_out + (long)BB * NSEL;           // [B, N]

    // workspace: z scores | selected-index ints | repacked W1 fragments
    float* z      = (float*)d_ws;                               // 512 KB
    int*   selIdx = (int*)((char*)d_ws + (size_t)MTOT * 4);     // 128 KB
    float* Wp     = (float*)((char*)d_ws + (size_t)MTOT * 4
                                         + (size_t)BB * NSEL * 4); // 128 KB

    // K0: repack W1 into B-fragment order (16384 float2 records)
    repack_W1_kernel<<<dim3(64), dim3(256), 0, stream>>>(W1, Wp);

    // K1: 4096 32-row wave-pairs, 8 waves/block
    mlp_logits_kernel<<<dim3(MTOT / 32 / 8), dim3(256), 0, stream>>>(
        features, Wp, b1, W2, b2, gumbel, log_out, z);

    // K2: 16*64 = 1024 (batch, parcel) waves, 8 waves/block
    topk_kernel<<<dim3(BB * NPAR / 8), dim3(256), 0, stream>>>(
        z, selIdx, idx_out);

    // K3: 16*2048 = 32768 row-gather waves, 8 waves/block
    gather_kernel<<<dim3(BB * NSEL / 8), dim3(256), 0, stream>>>(
        patches, selIdx, sel_out);
}